// Layer_46170898432476
// MI455X (gfx1250) — compile-verified
//
#include <hip/hip_runtime.h>
#include <math.h>

// ---------------------------------------------------------------------------
// Transformer layer for MI455X (gfx1250), exact-fp32 path via
// V_WMMA_F32_16X16X4_F32. Layer is HBM-bound (~350MB @ 23.3TB/s), so fp32
// matrix cores are more than sufficient and keep reference numerics.
// This revision adds CDNA5 async global->LDS staging (ASYNCcnt) with
// double-buffered GEMM tiles for copy/compute overlap.
// Workspace requirement: 4096*6144 + 2*4096*2048 floats = 167,772,160 bytes.
// ---------------------------------------------------------------------------

typedef __attribute__((ext_vector_type(2))) float v2f;
typedef __attribute__((ext_vector_type(8))) float v8f;

#define WMMA_F32(a, b, c) \
  __builtin_amdgcn_wmma_f32_16x16x4_f32(false, (a), false, (b), (short)0, (c), false, false)

#if defined(__has_builtin)
#if __has_builtin(__builtin_amdgcn_global_load_async_to_lds_b128)
#define HAVE_ASYNC_LDS 1
#endif
#endif

// Builtin parameter types per clang diagnostic: pointer to 16-byte int vector
// in AS(1) (global) / AS(3) (LDS).
typedef __attribute__((__vector_size__(4 * sizeof(int)))) int v4i_t;
typedef __attribute__((address_space(1))) v4i_t gv4i_t;
typedef __attribute__((address_space(3))) v4i_t lv4i_t;

// 16-byte global -> LDS copy; async DMA path on CDNA5, float4 fallback.
__device__ __forceinline__ void copy16_g2l(const float* __restrict__ g,
                                           float* __restrict__ l) {
#ifdef HAVE_ASYNC_LDS
  __builtin_amdgcn_global_load_async_to_lds_b128((gv4i_t*)g, (lv4i_t*)l, 0, 0);
#else
  *(float4*)l = *(const float4*)g;
#endif
}

__device__ __forceinline__ void wait_async_lds() {
#ifdef HAVE_ASYNC_LDS
#if __has_builtin(__builtin_amdgcn_s_wait_asynccnt)
  __builtin_amdgcn_s_wait_asynccnt(0);
#else
  asm volatile("s_wait_asynccnt 0x0" ::: "memory");
#endif
#endif
}

__device__ __forceinline__ float gelu_exact(float x) {
  return 0.5f * x * (1.0f + erff(x * 0.70710678118654752440f));
}

// ---------------------------------------------------------------------------
// Generic fp32 WMMA GEMM: out = A[MxK] @ W[KxN'] (+bias, +res, or GLU).
// Block tile 128x64, 8 waves (wave32) in 4x2, each wave 32x32 (2x2 WMMA tiles).
// LDS tiles are double-buffered and filled with async global->LDS copies so
// the DMA for tile kt+1 overlaps the 16 WMMAs of tile kt.
// EPI: 0 = +bias; 1 = +bias+res; 2 = (A@Wv+bv) * gelu(A@Wg+bg)  (w1 fusion).
// ---------------------------------------------------------------------------
template <int EPI>
__global__ __launch_bounds__(256) void gemm_wmma_f32(
    const float* __restrict__ A, const float* __restrict__ W,
    const float* __restrict__ bias, const float* __restrict__ res,
    float* __restrict__ out, int M, int N, int K, int ldw, int out_ld,
    int gate_off) {
  (void)M; (void)N;
  __shared__ float As[2][128][20];  // 128x16 A tile, stride 20 keeps 16B rows
  __shared__ float Ws[2][16][68];   // 16x64 W tile (value half for EPI==2)
  __shared__ float Wg[2][16][68];   // 16x64 gate tile (EPI==2 only)

  const int tid  = threadIdx.x;
  const int lane = tid & 31, wv = tid >> 5;
  const int wm = wv >> 1, wn = wv & 1;
  const int hi = lane >> 4, ln = lane & 15;
  const int m0 = blockIdx.y * 128;
  const int n0 = blockIdx.x * 64;

  v8f acc[2][2] = {};
  v8f agc[2][2] = {};

  const int nK = K >> 4;

  // issue async copies for K-tile kt into LDS buffer `buf`
  auto issue_tile = [&](int kt, int buf) {
    const int kb = kt << 4;
#pragma unroll
    for (int i = 0; i < 2; ++i) {
      const int idx = tid + (i << 8);
      const int r = idx >> 2, c = (idx & 3) << 2;
      copy16_g2l(&A[(size_t)(m0 + r) * K + kb + c], &As[buf][r][c]);
    }
    const int r = tid >> 4, c = (tid & 15) << 2;
    const float* wsrc = &W[(size_t)(kb + r) * ldw + n0 + c];
    copy16_g2l(wsrc, &Ws[buf][r][c]);
    if (EPI == 2) copy16_g2l(wsrc + gate_off, &Wg[buf][r][c]);
  };

  issue_tile(0, 0);

  for (int kt = 0; kt < nK; ++kt) {
    wait_async_lds();   // this wave's copies for tile kt have landed
    __syncthreads();    // ... and everyone else's too; prev compute done
    if (kt + 1 < nK) issue_tile(kt + 1, (kt + 1) & 1);
    const int buf = kt & 1;

    // ---- 4 WMMA k-steps over the 16-deep LDS tile ----
#pragma unroll
    for (int kk = 0; kk < 16; kk += 4) {
      const int kc = kk + 2 * hi;  // A/B frag: VGPR0 holds K=2*hi, VGPR1 K=2*hi+1
      v2f a[2], b[2], bg[2];
#pragma unroll
      for (int i = 0; i < 2; ++i) {
        const int r = wm * 32 + i * 16 + ln;
        a[i].x = As[buf][r][kc];
        a[i].y = As[buf][r][kc + 1];
      }
#pragma unroll
      for (int j = 0; j < 2; ++j) {
        const int cI = wn * 32 + j * 16 + ln;
        b[j].x = Ws[buf][kc][cI];
        b[j].y = Ws[buf][kc + 1][cI];
        if (EPI == 2) {
          bg[j].x = Wg[buf][kc][cI];
          bg[j].y = Wg[buf][kc + 1][cI];
        }
      }
#pragma unroll
      for (int i = 0; i < 2; ++i)
#pragma unroll
        for (int j = 0; j < 2; ++j) {
          acc[i][j] = WMMA_F32(a[i], b[j], acc[i][j]);
          if (EPI == 2) agc[i][j] = WMMA_F32(a[i], bg[j], agc[i][j]);
        }
    }
  }

  // ---- epilogue: C layout row = v + 8*hi, col = ln ----
#pragma unroll
  for (int i = 0; i < 2; ++i)
#pragma unroll
    for (int j = 0; j < 2; ++j)
#pragma unroll
      for (int v = 0; v < 8; ++v) {
        const int gr = m0 + wm * 32 + i * 16 + v + 8 * hi;
        const int gc = n0 + wn * 32 + j * 16 + ln;
        float val = acc[i][j][v] + bias[gc];
        if (EPI == 1) val += res[(size_t)gr * out_ld + gc];
        if (EPI == 2) val *= gelu_exact(agc[i][j][v] + bias[gc + gate_off]);
        out[(size_t)gr * out_ld + gc] = val;
      }
}

// ---------------------------------------------------------------------------
// RoPE, faithful to the reference (rot*cos + roll(rot,1)*sin), in-place on the
// q and k thirds of the qkv buffer. One wave per 128-dim head vector; the
// vector is staged in LDS so cross-element reads are race-free.
// ---------------------------------------------------------------------------
__global__ __launch_bounds__(256) void rope_kernel(float* __restrict__ qkv) {
  constexpr int S = 2048, D = 2048;
  __shared__ float rb[8][128];
  const int tid = threadIdx.x, lane = tid & 31, wv = tid >> 5;
  const int vid = blockIdx.x * 8 + wv;        // (bs, h, which) packed
  const int which = vid & 1;                  // 0 = q, 1 = k
  const int h = (vid >> 1) & 15;
  const int bs = vid >> 5;                    // b*S + s
  const int s = bs & (S - 1);
  float* base = qkv + (size_t)bs * (3 * D) + which * D + h * 128;

#pragma unroll
  for (int t = 0; t < 4; ++t) rb[wv][lane + 32 * t] = base[lane + 32 * t];
  // same-wave DS ops are in order; all stores precede all loads below

#pragma unroll
  for (int t = 0; t < 4; ++t) {
    const int j = lane + 32 * t;
    const float inv = powf(10000.0f, -(float)(j & 63) * (1.0f / 64.0f));
    const float ang = (float)s * inv;
    const float c = cosf(ang), sn = sinf(ang);
    float o;
    if (j & 1) {
      o = rb[wv][j - 1] * c - rb[wv][j] * sn;
    } else {
      const float prev = (j == 0) ? rb[wv][126] : rb[wv][j - 2];
      o = -rb[wv][j + 1] * c + prev * sn;
    }
    base[j] = o;
  }
}

// ---------------------------------------------------------------------------
// Flash attention (online softmax), one block per (128 q-rows, b, h).
// Avoids the 537MB score tensor that would thrash the 192MB L2. Q fragments
// live in registers; K/V tiles (64x128) staged in LDS via async copies; P is
// converted from C to A layout through a per-wave LDS buffer.
// ---------------------------------------------------------------------------
__global__ __launch_bounds__(256) void flash_attn_kernel(
    const float* __restrict__ qkv, const int* __restrict__ amask,
    float* __restrict__ out) {
  constexpr int S = 2048, D = 2048;
  constexpr int QLD = 3 * D;
  __shared__ float Ks[64][132];
  __shared__ float Vs[64][132];
  __shared__ float Ps[8][16][68];
  __shared__ int Ms[64];

  const int tid = threadIdx.x, lane = tid & 31, wv = tid >> 5;
  const int hi = lane >> 4, ln = lane & 15;
  const int bh = blockIdx.y, b = bh >> 4, h = bh & 15;
  const int q0 = blockIdx.x * 128;
  const float scale = 0.08838834764831845f;  // 1/sqrt(128)

  // Q A-fragments for this wave's 16 q-rows (row = ln), pre-scaled
  v2f aq[32];
  {
    const float* qr =
        qkv + (size_t)(b * S + q0 + wv * 16 + ln) * QLD + h * 128;
#pragma unroll
    for (int i = 0; i < 32; ++i) {
      const int c = 4 * i + 2 * hi;
      aq[i].x = qr[c] * scale;
      aq[i].y = qr[c + 1] * scale;
    }
  }

  v8f O[8] = {};
  float mrow[8], lrow[8];
#pragma unroll
  for (int v = 0; v < 8; ++v) { mrow[v] = -INFINITY; lrow[v] = 0.f; }

  for (int kt = 0; kt < S / 64; ++kt) {
    __syncthreads();  // all waves done with previous Ks/Vs
    {
      const float* kbp = qkv + (size_t)(b * S + kt * 64) * QLD + D + h * 128;
      const float* vbp = kbp + D;
#pragma unroll
      for (int t = 0; t < 8; ++t) {
        const int idx = (t << 8) + tid;
        const int r = idx >> 5, c4 = (idx & 31) << 2;
        copy16_g2l(kbp + (size_t)r * QLD + c4, &Ks[r][c4]);
        copy16_g2l(vbp + (size_t)r * QLD + c4, &Vs[r][c4]);
      }
      if (tid < 64) Ms[tid] = amask[b * S + kt * 64 + tid];
    }
    wait_async_lds();
    __syncthreads();

    // scores: wave computes 16 q-rows x 64 keys, K^T frags from Ks rows
    v8f sacc[4] = {};
#pragma unroll
    for (int kk = 0; kk < 128; kk += 4) {
      const int kc = kk + 2 * hi;
      const v2f a = aq[kk >> 2];
#pragma unroll
      for (int t = 0; t < 4; ++t) {
        v2f bf;
        bf.x = Ks[t * 16 + ln][kc];
        bf.y = Ks[t * 16 + ln][kc + 1];
        sacc[t] = WMMA_F32(a, bf, sacc[t]);
      }
    }

    // online softmax update (per C-layout row v + 8*hi)
#pragma unroll
    for (int v = 0; v < 8; ++v) {
      float mx = -INFINITY;
#pragma unroll
      for (int t = 0; t < 4; ++t) {
        float sv = sacc[t][v];
        if (Ms[t * 16 + ln] == 0) sv = -INFINITY;
        sacc[t][v] = sv;
        mx = fmaxf(mx, sv);
      }
#pragma unroll
      for (int off = 1; off < 16; off <<= 1)
        mx = fmaxf(mx, __shfl_xor(mx, off, 32));
      const float mnew = fmaxf(mrow[v], mx);
      float ssum = 0.f;
#pragma unroll
      for (int t = 0; t < 4; ++t) {
        const float sv = sacc[t][v];
        const float p = (sv == -INFINITY) ? 0.f : expf(sv - mnew);
        sacc[t][v] = p;
        ssum += p;
      }
#pragma unroll
      for (int off = 1; off < 16; off <<= 1) ssum += __shfl_xor(ssum, off, 32);
      const float corr = (mrow[v] == -INFINITY) ? 0.f : expf(mrow[v] - mnew);
      lrow[v] = lrow[v] * corr + ssum;
      mrow[v] = mnew;
#pragma unroll
      for (int t = 0; t < 8; ++t) O[t][v] *= corr;
    }

    // stash P into per-wave LDS (C layout -> memory), then read as A frags.
    // Same-wave LDS ops execute in order; no barrier needed.
#pragma unroll
    for (int t = 0; t < 4; ++t)
#pragma unroll
      for (int v = 0; v < 8; ++v)
        Ps[wv][v + 8 * hi][t * 16 + ln] = sacc[t][v];

    // O += P @ V
#pragma unroll
    for (int kk = 0; kk < 64; kk += 4) {
      const int kc = kk + 2 * hi;
      v2f a;
      a.x = Ps[wv][ln][kc];
      a.y = Ps[wv][ln][kc + 1];
#pragma unroll
      for (int t = 0; t < 8; ++t) {
        v2f bf;
        bf.x = Vs[kc][t * 16 + ln];
        bf.y = Vs[kc + 1][t * 16 + ln];
        O[t] = WMMA_F32(a, bf, O[t]);
      }
    }
  }

  // normalized output -> [B,S,D] with head columns at h*128
#pragma unroll
  for (int v = 0; v < 8; ++v) {
    const float inv = (lrow[v] > 0.f) ? 1.0f / lrow[v] : 0.f;
    const int row = q0 + wv * 16 + v + 8 * hi;
#pragma unroll
    for (int t = 0; t < 8; ++t)
      out[(size_t)(b * S + row) * D + h * 128 + t * 16 + ln] = O[t][v] * inv;
  }
}

// ---------------------------------------------------------------------------
// In-place RMSNorm over rows of length D=2048 (block per row).
// ---------------------------------------------------------------------------
__global__ __launch_bounds__(256) void rmsnorm_kernel(
    float* __restrict__ x, const float* __restrict__ w) {
  constexpr int D = 2048;
  __shared__ float red[8];
  float* row = x + (size_t)blockIdx.x * D;
  const int tid = threadIdx.x;
  float vals[8];
  float ss = 0.f;
#pragma unroll
  for (int i = 0; i < 8; ++i) {
    vals[i] = row[tid + 256 * i];
    ss += vals[i] * vals[i];
  }
#pragma unroll
  for (int off = 16; off > 0; off >>= 1) ss += __shfl_xor(ss, off, 32);
  if ((tid & 31) == 0) red[tid >> 5] = ss;
  __syncthreads();
  float tot = 0.f;
#pragma unroll
  for (int i = 0; i < 8; ++i) tot += red[i];
  const float inv = 1.0f / sqrtf(tot * (1.0f / D) + 1e-6f);
#pragma unroll
  for (int i = 0; i < 8; ++i) {
    const int c = tid + 256 * i;
    row[c] = w[c] * vals[i] * inv;
  }
}

// ---------------------------------------------------------------------------
extern "C" void kernel_launch(void* const* d_in, const int* in_sizes, int n_in,
                              void* d_out, int out_size, void* d_ws,
                              size_t ws_size, hipStream_t stream) {
  (void)in_sizes; (void)n_in; (void)out_size; (void)ws_size;
  constexpr int Bx = 2, S = 2048, D = 2048;
  constexpr int M = Bx * S;  // 4096

  const float* x     = (const float*)d_in[0];
  const int*   amask = (const int*)d_in[1];
  const float* w_qkv = (const float*)d_in[2];
  const float* b_qkv = (const float*)d_in[3];
  const float* w_o   = (const float*)d_in[4];
  const float* b_o   = (const float*)d_in[5];
  const float* ln1_w = (const float*)d_in[6];
  const float* w1    = (const float*)d_in[7];
  const float* b1    = (const float*)d_in[8];
  const float* w2    = (const float*)d_in[9];
  const float* b2    = (const float*)d_in[10];
  const float* ln2_w = (const float*)d_in[11];
  float* out = (float*)d_out;

  // Workspace layout (floats), with aliasing:
  //   [0,              M*3D)  qkv         -- dead after attention
  //   [M*3D,    M*3D + M*D)   attn out    -- dead after o-proj
  //   [M*3D+M*D, M*3D+2*M*D)  x1 (y->norm in place)
  //   hbuf = [0, M*4D)        FFN hidden  -- reuses qkv+attn region
  float* ws   = (float*)d_ws;
  float* qkv  = ws;
  float* attn = ws + (size_t)M * 3 * D;
  float* x1   = attn + (size_t)M * D;
  float* hbuf = ws;

  // 1) qkv = x @ w_qkv + b_qkv
  gemm_wmma_f32<0><<<dim3(3 * D / 64, M / 128), 256, 0, stream>>>(
      x, w_qkv, b_qkv, nullptr, qkv, M, 3 * D, D, 3 * D, 3 * D, 0);
  // 2) RoPE in place on q,k (B*S*H*2 = 131072 head-vectors, 8 per block)
  rope_kernel<<<dim3(M * 16 * 2 / 8), 256, 0, stream>>>(qkv);
  // 3) flash attention -> attn [B,S,D]
  flash_attn_kernel<<<dim3(S / 128, Bx * 16), 256, 0, stream>>>(qkv, amask,
                                                                attn);
  // 4) y = attn @ w_o + b_o + x  (residual fused), then 5) RMSNorm in place
  gemm_wmma_f32<1><<<dim3(D / 64, M / 128), 256, 0, stream>>>(
      attn, w_o, b_o, x, x1, M, D, D, D, D, 0);
  rmsnorm_kernel<<<dim3(M), 256, 0, stream>>>(x1, ln1_w);
  // 6) h = (x1@w1_val + b) * gelu(x1@w1_gate + b)   (fused GLU, writes 4D)
  gemm_wmma_f32<2><<<dim3(4 * D / 64, M / 128), 256, 0, stream>>>(
      x1, w1, b1, nullptr, hbuf, M, 4 * D, D, 8 * D, 4 * D, 4 * D);
  // 7) out = h @ w2 + b2 + x1, then 8) RMSNorm in place on d_out
  gemm_wmma_f32<1><<<dim3(D / 64, M / 128), 256, 0, stream>>>(
      hbuf, w2, b2, x1, out, M, D, 4 * D, D, D, 0);
  rmsnorm_kernel<<<dim3(M), 256, 0, stream>>>(out, ln2_w);
}